// HyperbolicGINLayer_57638461112980
// MI455X (gfx1250) — compile-verified
//
#include <hip/hip_runtime.h>
#include <hip/hip_bf16.h>

#define DIM 128

typedef float v2f __attribute__((ext_vector_type(2)));
typedef float v8f __attribute__((ext_vector_type(8)));

// ---------------------------------------------------------------------------
// wave32 reduction helpers
// ---------------------------------------------------------------------------
__device__ __forceinline__ float wred_add(float v) {
#pragma unroll
  for (int m = 16; m > 0; m >>= 1) v += __shfl_xor(v, m);
  return v;
}

// ---------------------------------------------------------------------------
// Pack 127x127 weight into WMMA-B-native pair layout.
// Logical Wt[k][j] = w[j-1][k-1] (row0/col0 = 0), k,j in [0,128).
// Pair index p = (g*2 + h)*128 + j  (g = k-group k0/4, h = half-wave)
//   Wp[2p+0] = Wt[4g+2h+0][j],  Wp[2p+1] = Wt[4g+2h+1][j]
// -> in the GEMM, lane (j, half h) reads ONE float2 per K-step, coalesced.
// Bias padded: bp[0]=0, bp[j]=b[j-1].
// ---------------------------------------------------------------------------
__global__ __launch_bounds__(256) void pad_weight_kernel(
    const float* __restrict__ w, const float* __restrict__ b,
    float* __restrict__ Wp, float* __restrict__ bp) {
  int p = blockIdx.x * 256 + threadIdx.x;        // 0 .. 16383 (pair index)
  int j = p & 127;
  int q = p >> 7;                                // g*2 + h
  int k = 2 * q;                                 // = 4g + 2h, even, 0..254
  float v0 = 0.0f, v1 = 0.0f;
  if (j >= 1) {
    if (k >= 1) v0 = w[(j - 1) * 127 + (k - 1)];
    v1 = w[(j - 1) * 127 + k];                   // no bound check: row k+1 reaches 255
  }
  ((v2f*)Wp)[p] = (v2f){v0, v1};
  if (p < DIM) bp[p] = (p == 0) ? 0.0f : b[p - 1];
}

// ---------------------------------------------------------------------------
// WMMA fp32 GEMM:  Out[n][j] = sum_k A[n][k] * Wt[k][j] + bp[j]
// One block = 16 rows of A; 8 waves each own a 16-wide column tile.
// A tile staged in LDS via async global->LDS (ASYNCcnt), stride 132.
// ---------------------------------------------------------------------------
#define A_STRIDE 132

__global__ __launch_bounds__(256) void gemm128_wmma_kernel(
    const float* __restrict__ A, const float* __restrict__ Wp,
    const float* __restrict__ bp, float* __restrict__ Out, int nrows) {
  __shared__ float aTile[16 * A_STRIDE];
  const int tid = threadIdx.x;
  const int wid = tid >> 5;
  const int lane = tid & 31;
  const int nb = blockIdx.x * 16;

  // cooperative stage of A tile (16 x 128 f32): async global->LDS b128
#pragma unroll
  for (int i = 0; i < 2; ++i) {
    int pos = i * 256 + tid;      // float4 index, 32 per row
    int row = pos >> 5;
    int col = (pos & 31) << 2;
    float* ldst = &aTile[row * A_STRIDE + col];
    if (nb + row < nrows) {
      const float* gsrc = A + (size_t)(nb + row) * DIM + col;
      unsigned lds_off = (unsigned)(uintptr_t)ldst;   // flat LDS: offset in addr[31:0]
      asm volatile("global_load_async_to_lds_b128 %0, %1, off"
                   :: "v"(lds_off), "v"(gsrc)
                   : "memory");
    } else {
      *(float4*)ldst = make_float4(0.f, 0.f, 0.f, 0.f);
    }
  }
  asm volatile("s_wait_asynccnt 0x0" ::: "memory");
  __syncthreads();

  const int jb = wid * 16;                  // this wave's output column base
  const int m = lane & 15;
  const int half = lane >> 4;               // 0: K rows +0/+1, 1: +2/+3
  const int kb = half << 1;

  v8f acc;
  const float bias = bp[jb + m];
#pragma unroll
  for (int r = 0; r < 8; ++r) acc[r] = bias;

  const v2f* WpV = (const v2f*)Wp;
#pragma unroll
  for (int k0 = 0; k0 < DIM; k0 += 4) {
    // A 16x4: lane m (half h): VGPR0 = A[m][k0+2h], VGPR1 = A[m][k0+2h+1]
    v2f a = *(const v2f*)(&aTile[m * A_STRIDE + k0 + kb]);
    // B 4x16 pair, single coalesced b64 load per lane
    v2f b = WpV[(size_t)(((k0 >> 2) * 2 + half) * DIM) + jb + m];
    acc = __builtin_amdgcn_wmma_f32_16x16x4_f32(false, a, false, b,
                                                (short)0, acc, false, false);
  }

  // D layout: VGPR r -> M = r + 8*half, N = lane%16
  const int mbase = nb + (half << 3);
#pragma unroll
  for (int r = 0; r < 8; ++r) {
    int row = mbase + r;
    if (row < nrows) Out[(size_t)row * DIM + jb + m] = acc[r];
  }
}

// ---------------------------------------------------------------------------
// logmap0: in = hyperbolic point [x0, xs...], out = tangent [0, d*xs/|xs|]
// One wave per node; each lane owns 4 components.
// ---------------------------------------------------------------------------
__global__ __launch_bounds__(256) void logmap0_kernel(
    const float* __restrict__ in, float* __restrict__ out, int n) {
  const int lane = threadIdx.x & 31;
  const int node = (blockIdx.x * blockDim.x + threadIdx.x) >> 5;
  if (node >= n) return;
  const float4 v = *(const float4*)(in + (size_t)node * DIM + lane * 4);
  float ss = v.x * v.x + v.y * v.y + v.z * v.z + v.w * v.w;
  if (lane == 0) ss -= v.x * v.x;          // exclude time component
  ss = wred_add(ss);
  const float x0 = __shfl(v.x, 0);
  const float xn = fmaxf(sqrtf(ss), 1e-9f);
  const float d = acoshf(fmaxf(x0, 1.0f + 1e-7f));
  const float f = d / xn;
  float4 o;
  o.x = (lane == 0) ? 0.0f : v.x * f;
  o.y = v.y * f; o.z = v.z * f; o.w = v.w * f;
  *(float4*)(out + (size_t)node * DIM + lane * 4) = o;
}

// ---------------------------------------------------------------------------
// proj(expmap0(tangent)): in comp0 == 0.  out = [sqrt(1+|xs|^2), xs]
// ---------------------------------------------------------------------------
__global__ __launch_bounds__(256) void expmap_proj_kernel(
    const float* __restrict__ in, float* __restrict__ out, int n) {
  const int lane = threadIdx.x & 31;
  const int node = (blockIdx.x * blockDim.x + threadIdx.x) >> 5;
  if (node >= n) return;
  const float4 v = *(const float4*)(in + (size_t)node * DIM + lane * 4);
  float ss = v.x * v.x + v.y * v.y + v.z * v.z + v.w * v.w;
  ss = wred_add(ss);
  const float un = sqrtf(ss);
  const float unc = fmaxf(un, 1e-9f);
  const float f = sinhf(unc) / unc;
  const float x0 = sqrtf(1.0f + f * f * ss);
  float4 o;
  o.x = (lane == 0) ? x0 : v.x * f;
  o.y = v.y * f; o.z = v.z * f; o.w = v.w * f;
  *(float4*)(out + (size_t)node * DIM + lane * 4) = o;
}

// ---------------------------------------------------------------------------
// Per-edge attention + fused scatter:  s[row] += exp(-dist(xl[row],xl[col])) * x[col]
// One wave per edge; L2-resident gathers + global f32 atomics.
// ---------------------------------------------------------------------------
__global__ __launch_bounds__(256) void edge_att_scatter_kernel(
    const float* __restrict__ xl, const float* __restrict__ x,
    const int* __restrict__ ei, float* __restrict__ s, int nE) {
  const int lane = threadIdx.x & 31;
  const int e = (blockIdx.x * blockDim.x + threadIdx.x) >> 5;
  if (e >= nE) return;
  const int r = ei[e];
  const int c = ei[nE + e];
  const float4 a = *(const float4*)(xl + (size_t)r * DIM + lane * 4);
  const float4 b = *(const float4*)(xl + (size_t)c * DIM + lane * 4);
  float d = a.x * b.x + a.y * b.y + a.z * b.z + a.w * b.w;
  if (lane == 0) d -= 2.0f * a.x * b.x;    // Lorentz: -x0*y0 + sum xs*ys
  const float li = wred_add(d);
  const float u = fmaxf(-li, 1.0f + 1e-7f);
  const float att = expf(-acoshf(u));
  const float4 xc = *(const float4*)(x + (size_t)c * DIM + lane * 4);
  float* dst = s + (size_t)r * DIM + lane * 4;
  atomicAdd(dst + 0, att * xc.x);
  atomicAdd(dst + 1, att * xc.y);
  atomicAdd(dst + 2, att * xc.z);
  atomicAdd(dst + 3, att * xc.w);
}

// ---------------------------------------------------------------------------
// Combine: agg = proj(s/denom); v = logmap0(agg) + (1+eps)*logmap0(proj(x));
// out = proj(expmap0(v))
// ---------------------------------------------------------------------------
__global__ __launch_bounds__(256) void combine_kernel(
    const float* __restrict__ s, const float* __restrict__ x,
    const float* __restrict__ eps_p, float* __restrict__ out, int n) {
  const int lane = threadIdx.x & 31;
  const int node = (blockIdx.x * blockDim.x + threadIdx.x) >> 5;
  if (node >= n) return;
  const float4 sv = *(const float4*)(s + (size_t)node * DIM + lane * 4);
  const float4 xv = *(const float4*)(x + (size_t)node * DIM + lane * 4);

  float ls = sv.x * sv.x + sv.y * sv.y + sv.z * sv.z + sv.w * sv.w;
  if (lane == 0) ls -= 2.0f * sv.x * sv.x;               // Lorentz inner(s,s)
  float xs = xv.y * xv.y + xv.z * xv.z + xv.w * xv.w;
  if (lane != 0) xs += xv.x * xv.x;                      // spatial |x|^2
  const float li = wred_add(ls);
  const float xsp = wred_add(xs);

  const float s0 = __shfl(sv.x, 0);
  const float sp = li + s0 * s0;                         // spatial |s|^2
  const float denom = sqrtf(fmaxf(fabsf(li), 1e-9f));

  // logmap0(proj(s/denom))
  const float agg0 = sqrtf(1.0f + sp / (denom * denom));
  const float xn1 = fmaxf(sqrtf(sp) / denom, 1e-9f);
  const float d1 = acoshf(fmaxf(agg0, 1.0f + 1e-7f));
  const float fa = d1 / (xn1 * denom);                   // applied to s_k

  // (1+eps)*logmap0(proj(x))
  const float epsv = *eps_p;
  const float xp0 = sqrtf(1.0f + xsp);
  const float xn2 = fmaxf(sqrtf(xsp), 1e-9f);
  const float d2 = acoshf(fmaxf(xp0, 1.0f + 1e-7f));
  const float fx = (d2 / xn2) * (1.0f + epsv);

  float4 vv;
  vv.x = (lane == 0) ? 0.0f : (sv.x * fa + xv.x * fx);
  vv.y = sv.y * fa + xv.y * fx;
  vv.z = sv.z * fa + xv.z * fx;
  vv.w = sv.w * fa + xv.w * fx;

  float vsq = vv.x * vv.x + vv.y * vv.y + vv.z * vv.z + vv.w * vv.w;
  vsq = wred_add(vsq);
  const float un = sqrtf(vsq);
  const float unc = fmaxf(un, 1e-9f);
  const float f = sinhf(unc) / unc;
  const float h0 = sqrtf(1.0f + f * f * vsq);
  float4 o;
  o.x = (lane == 0) ? h0 : vv.x * f;
  o.y = vv.y * f; o.z = vv.z * f; o.w = vv.w * f;
  *(float4*)(out + (size_t)node * DIM + lane * 4) = o;
}

// ---------------------------------------------------------------------------
// Fused post-GEMM chain: t = proj(expmap0([0,os])); out = expmap0([0, tanh(logmap0(t)[1:])])
// ---------------------------------------------------------------------------
__global__ __launch_bounds__(256) void post_act_kernel(
    const float* __restrict__ os, float* __restrict__ out, int n) {
  const int lane = threadIdx.x & 31;
  const int node = (blockIdx.x * blockDim.x + threadIdx.x) >> 5;
  if (node >= n) return;
  const float4 v = *(const float4*)(os + (size_t)node * DIM + lane * 4);
  float ss = v.x * v.x + v.y * v.y + v.z * v.z + v.w * v.w;  // comp0 == 0
  ss = wred_add(ss);
  const float un1 = sqrtf(ss);
  const float u1c = fmaxf(un1, 1e-9f);
  const float f1 = sinhf(u1c) / u1c;
  const float t0 = sqrtf(1.0f + f1 * f1 * ss);
  // logmap0(t): v_k = d * (f1*os_k) / |f1*os|
  const float tn = fmaxf(f1 * un1, 1e-9f);
  const float d = acoshf(fmaxf(t0, 1.0f + 1e-7f));
  const float g = d * f1 / tn;
  float4 tv;
  tv.x = tanhf(v.x * g);
  tv.y = tanhf(v.y * g);
  tv.z = tanhf(v.z * g);
  tv.w = tanhf(v.w * g);
  float vs2 = tv.x * tv.x + tv.y * tv.y + tv.z * tv.z + tv.w * tv.w;
  vs2 = wred_add(vs2);
  const float un2 = fmaxf(sqrtf(vs2), 1e-9f);
  const float f2 = sinhf(un2) / un2;
  float4 o;
  o.x = (lane == 0) ? coshf(un2) : tv.x * f2;
  o.y = tv.y * f2; o.z = tv.z * f2; o.w = tv.w * f2;
  *(float4*)(out + (size_t)node * DIM + lane * 4) = o;
}

// ---------------------------------------------------------------------------
extern "C" void kernel_launch(void* const* d_in, const int* in_sizes, int n_in,
                              void* d_out, int out_size, void* d_ws, size_t ws_size,
                              hipStream_t stream) {
  const float* x     = (const float*)d_in[0];
  const int*   ei    = (const int*)d_in[1];
  const float* att_w = (const float*)d_in[2];
  const float* att_b = (const float*)d_in[3];
  const float* w1    = (const float*)d_in[4];
  const float* b1    = (const float*)d_in[5];
  const float* w2    = (const float*)d_in[6];
  const float* b2    = (const float*)d_in[7];
  const float* eps   = (const float*)d_in[8];
  float* out = (float*)d_out;

  const int N = in_sizes[0] / DIM;
  const int E = in_sizes[1] / 2;

  float* ws = (float*)d_ws;
  float* W0 = ws;                    // 128*128 each (packed pair layout)
  float* W1 = W0 + DIM * DIM;
  float* W2 = W1 + DIM * DIM;
  float* B0 = W2 + DIM * DIM;        // 128 each
  float* B1 = B0 + DIM;
  float* B2 = B1 + DIM;
  float* bufA = B2 + DIM;            // N*128 each
  float* bufB = bufA + (size_t)N * DIM;
  float* bufC = bufB + (size_t)N * DIM;

  const int nodeBlocks = (N + 7) / 8;     // wave-per-node kernels
  const int edgeBlocks = (E + 7) / 8;     // wave-per-edge kernel
  const int gemmBlocks = (N + 15) / 16;

  // 0) pack weights + biases into WMMA-friendly layout
  pad_weight_kernel<<<64, 256, 0, stream>>>(att_w, att_b, W0, B0);
  pad_weight_kernel<<<64, 256, 0, stream>>>(w1, b1, W1, B1);
  pad_weight_kernel<<<64, 256, 0, stream>>>(w2, b2, W2, B2);

  // 1) attention linear: xl = proj(expmap0(logmap0(x) @ Wt + b))
  logmap0_kernel<<<nodeBlocks, 256, 0, stream>>>(x, bufA, N);
  gemm128_wmma_kernel<<<gemmBlocks, 256, 0, stream>>>(bufA, W0, B0, bufB, N);
  hipMemsetAsync(bufA, 0, (size_t)N * DIM * sizeof(float), stream);   // s = 0
  expmap_proj_kernel<<<nodeBlocks, 256, 0, stream>>>(bufB, bufC, N);  // xl

  // 2) edge attention + segment_sum scatter into s (= bufA)
  edge_att_scatter_kernel<<<edgeBlocks, 256, 0, stream>>>(bufC, x, ei, bufA, E);

  // 3) aggregation + residual combine -> h_pre (bufB)
  combine_kernel<<<nodeBlocks, 256, 0, stream>>>(bufA, x, eps, bufB, N);

  // 4) h = hyp_act(hyp_linear(h, w1, b1))
  logmap0_kernel<<<nodeBlocks, 256, 0, stream>>>(bufB, bufC, N);
  gemm128_wmma_kernel<<<gemmBlocks, 256, 0, stream>>>(bufC, W1, B1, bufA, N);
  post_act_kernel<<<nodeBlocks, 256, 0, stream>>>(bufA, bufB, N);

  // 5) h = hyp_act(hyp_linear(h, w2, b2)) -> d_out
  logmap0_kernel<<<nodeBlocks, 256, 0, stream>>>(bufB, bufC, N);
  gemm128_wmma_kernel<<<gemmBlocks, 256, 0, stream>>>(bufC, W2, B2, bufA, N);
  post_act_kernel<<<nodeBlocks, 256, 0, stream>>>(bufA, out, N);
}